// HierarchicalGNN_56899726737799
// MI455X (gfx1250) — compile-verified
//
#include <hip/hip_runtime.h>

typedef __attribute__((ext_vector_type(2))) float v2f;
typedef __attribute__((ext_vector_type(8))) float v8f;

#define NEG_SLOPE 0.2f
#define GAT_EPS 1e-16f

// ---------------------------------------------------------------------------
// Order-preserving float <-> uint mapping for atomicMax on floats.
// ---------------------------------------------------------------------------
__device__ __forceinline__ unsigned f2ord(float f) {
  unsigned u = __float_as_uint(f);
  return (u & 0x80000000u) ? ~u : (u | 0x80000000u);
}
__device__ __forceinline__ float ord2f(unsigned u) {
  return (u & 0x80000000u) ? __uint_as_float(u ^ 0x80000000u)
                           : __uint_as_float(~u);
}

// ---------------------------------------------------------------------------
// WMMA f32 GEMM: C[M,N] = A[M,K] @ B[K,N]  (row-major, M%16==0, K%4==0,
// N % (16*TN) == 0). One wave32 computes a 16 x (16*TN) strip of C with TN
// independent V_WMMA_F32_16X16X4_F32 accumulators. Per K-step: issue the A
// load + all TN B load-pairs first, then fire the TN independent WMMAs
// back-to-back (single loadcnt wait, XDL pipe stays busy).
// ---------------------------------------------------------------------------
template <int TN>
__global__ void wmma_gemm_f32(const float* __restrict__ A,
                              const float* __restrict__ B,
                              float* __restrict__ C, int M, int N, int K) {
  const int lane   = threadIdx.x & 31;
  const int wave   = (int)((blockIdx.x * blockDim.x + threadIdx.x) >> 5);
  const int strips = (M >> 4) * (N / (16 * TN));
  if (wave >= strips) return;                // wave-uniform: EXEC stays all-1s
  const int stripsN = N / (16 * TN);
  const int tm = wave / stripsN;
  const int sn = wave - tm * stripsN;
  const int r  = lane & 15;
  const int kh = (lane >> 4) << 1;           // K sub-pair: 0 or 2 per half-wave

  const float* __restrict__ ap = A + (size_t)(tm * 16 + r) * K + kh;
  const float* __restrict__ bp = B + (size_t)kh * N + sn * (16 * TN) + r;

  v8f acc[TN];
  const v8f vzero = {0.f, 0.f, 0.f, 0.f, 0.f, 0.f, 0.f, 0.f};
#pragma unroll
  for (int j = 0; j < TN; ++j) acc[j] = vzero;

  for (int k = 0; k < K; k += 4) {
    __builtin_prefetch(ap + 16, 0, 3);       // near-scope speculative prefetch
    v2f a, b[TN];
    a.x = ap[0];
    a.y = ap[1];
#pragma unroll
    for (int j = 0; j < TN; ++j) {           // cluster all B loads
      b[j].x = bp[j * 16];
      b[j].y = bp[N + j * 16];
    }
#pragma unroll
    for (int j = 0; j < TN; ++j) {           // TN independent WMMAs in a row
      // 8 args: (neg_a, A, neg_b, B, c_mod, C, reuse_a, reuse_b)
      acc[j] = __builtin_amdgcn_wmma_f32_16x16x4_f32(false, a, false, b[j],
                                                     (short)0, acc[j], false,
                                                     false);
    }
    ap += 4;
    bp += (size_t)4 * N;
  }

  float* __restrict__ cp =
      C + (size_t)(tm * 16 + (lane >> 4) * 8) * N + sn * (16 * TN) + r;
#pragma unroll
  for (int j = 0; j < TN; ++j)
#pragma unroll
    for (int i = 0; i < 8; ++i) cp[(size_t)i * N + j * 16] = acc[j][i];
}

// ---------------------------------------------------------------------------
// Per-(node,head) attention logits: as/ad[n*H+h] = <h[n,h,:], a_src/dst[h,:]>
// H = 1<<Hs, C = 1<<Cs (always powers of two here).
// ---------------------------------------------------------------------------
__global__ void alpha_dots_k(const float* __restrict__ h,
                             const float* __restrict__ aS,
                             const float* __restrict__ aD,
                             float* __restrict__ as_, float* __restrict__ ad_,
                             int N, int Hs, int Cs) {
  int t = blockIdx.x * blockDim.x + threadIdx.x;
  if (t >= (N << Hs)) return;
  int hh = t & ((1 << Hs) - 1);
  int C = 1 << Cs;
  const float* __restrict__ hp = h + ((size_t)t << Cs);  // n*H*C + hh*C
  const float* __restrict__ sp = aS + ((size_t)hh << Cs);
  const float* __restrict__ dp = aD + ((size_t)hh << Cs);
  float vs = 0.f, vd = 0.f;
  for (int c = 0; c < C; ++c) {
    float x = hp[c];
    vs += x * sp[c];
    vd += x * dp[c];
  }
  as_[t] = vs;
  ad_[t] = vd;
}

// ---------------------------------------------------------------------------
// Generic u32 fill (init ordered-max buffers, sums, accumulators)
// ---------------------------------------------------------------------------
__global__ void fill_u32_k(unsigned* __restrict__ p, unsigned v, size_t n) {
  size_t t = (size_t)blockIdx.x * blockDim.x + threadIdx.x;
  if (t < n) p[t] = v;
}

// ---------------------------------------------------------------------------
// Edge helpers: edge i in [0,E) from edge_index, i in [E,E+N) = self loop.
// ---------------------------------------------------------------------------
__device__ __forceinline__ void edge_sd(const int* __restrict__ ei, int E,
                                        int i, int& s, int& d) {
  if (i < E) { s = ei[i]; d = ei[E + i]; } else { s = d = i - E; }
}
__device__ __forceinline__ float edge_logit(const float* __restrict__ as_,
                                            const float* __restrict__ ad_,
                                            int s, int d, int Hs, int hh) {
  float e = as_[((size_t)s << Hs) + hh] + ad_[((size_t)d << Hs) + hh];
  return e > 0.f ? e : NEG_SLOPE * e;
}

__global__ void edge_max_k(const int* __restrict__ ei, int E, int N, int Hs,
                           const float* __restrict__ as_,
                           const float* __restrict__ ad_,
                           unsigned* __restrict__ mo) {
  size_t t = (size_t)blockIdx.x * blockDim.x + threadIdx.x;
  size_t tot = (size_t)(E + N) << Hs;
  if (t >= tot) return;
  int i = (int)(t >> Hs), hh = (int)(t & ((1u << Hs) - 1));
  int s, d;
  edge_sd(ei, E, i, s, d);
  float e = edge_logit(as_, ad_, s, d, Hs, hh);
  atomicMax(&mo[((size_t)d << Hs) + hh], f2ord(e));
}

__global__ void edge_sum_k(const int* __restrict__ ei, int E, int N, int Hs,
                           const float* __restrict__ as_,
                           const float* __restrict__ ad_,
                           const unsigned* __restrict__ mo,
                           float* __restrict__ sb) {
  size_t t = (size_t)blockIdx.x * blockDim.x + threadIdx.x;
  size_t tot = (size_t)(E + N) << Hs;
  if (t >= tot) return;
  int i = (int)(t >> Hs), hh = (int)(t & ((1u << Hs) - 1));
  int s, d;
  edge_sd(ei, E, i, s, d);
  float e = edge_logit(as_, ad_, s, d, Hs, hh);
  size_t dh = ((size_t)d << Hs) + hh;
  atomicAdd(&sb[dh], __expf(e - ord2f(mo[dh])));
}

__global__ void edge_alpha_k(const int* __restrict__ ei, int E, int N, int Hs,
                             const float* __restrict__ as_,
                             const float* __restrict__ ad_,
                             const unsigned* __restrict__ mo,
                             const float* __restrict__ sb,
                             float* __restrict__ alphab) {
  size_t t = (size_t)blockIdx.x * blockDim.x + threadIdx.x;
  size_t tot = (size_t)(E + N) << Hs;
  if (t >= tot) return;
  int i = (int)(t >> Hs), hh = (int)(t & ((1u << Hs) - 1));
  int s, d;
  edge_sd(ei, E, i, s, d);
  float e = edge_logit(as_, ad_, s, d, Hs, hh);
  size_t dh = ((size_t)d << Hs) + hh;
  alphab[t] = __expf(e - ord2f(mo[dh])) / (sb[dh] + GAT_EPS);
}

// One thread per (edge,head,ch): out[dst,h,c] += alpha[e,h] * hbuf[src,h,c].
// Consecutive threads hit consecutive channels -> coalesced f32 atomics.
__global__ void edge_aggr_k(const int* __restrict__ ei, int E, int N, int Hs,
                            int Cs, const float* __restrict__ alphab,
                            const float* __restrict__ hbuf,
                            float* __restrict__ out) {
  size_t t = (size_t)blockIdx.x * blockDim.x + threadIdx.x;
  int Fs = Hs + Cs;
  size_t tot = (size_t)(E + N) << Fs;
  if (t >= tot) return;
  int i = (int)(t >> Fs);
  int rem = (int)(t & ((1u << Fs) - 1));
  int s, d;
  edge_sd(ei, E, i, s, d);
  float a = alphab[((size_t)i << Hs) + (rem >> Cs)];
  atomicAdd(&out[((size_t)d << Fs) + rem], a * hbuf[((size_t)s << Fs) + rem]);
}

__global__ void bias_act_k(float* __restrict__ out, const float* __restrict__ b,
                           int Fmask, size_t n, int relu) {
  size_t t = (size_t)blockIdx.x * blockDim.x + threadIdx.x;
  if (t >= n) return;
  float v = out[t] + b[(int)(t & (size_t)Fmask)];
  if (relu) v = fmaxf(v, 0.f);
  out[t] = v;
}

// e_in[n] = [ emp_x[n] (F1) | dept_emb[dept_idx[n]] (F2) ]
__global__ void concat_k(const float* __restrict__ emp_x,
                         const float* __restrict__ dvec,
                         const int* __restrict__ didx, float* __restrict__ out,
                         int N, int F1, int F2) {
  size_t t = (size_t)blockIdx.x * blockDim.x + threadIdx.x;
  int F = F1 + F2;
  size_t tot = (size_t)N * F;
  if (t >= tot) return;
  int n = (int)(t / F), j = (int)(t - (size_t)n * F);
  out[t] = (j < F1) ? emp_x[(size_t)n * F1 + j]
                    : dvec[(size_t)didx[n] * F2 + (j - F1)];
}

// log_softmax over rows of exactly 32 channels: one wave32 per row.
__global__ void log_softmax32_k(const float* __restrict__ in,
                                float* __restrict__ out, int N) {
  int lane = threadIdx.x & 31;
  int row = (int)((blockIdx.x * blockDim.x + threadIdx.x) >> 5);
  if (row >= N) return;
  float v = in[(size_t)row * 32 + lane];
  float m = v;
#pragma unroll
  for (int o = 16; o; o >>= 1) m = fmaxf(m, __shfl_xor(m, o, 32));
  float s = __expf(v - m);
#pragma unroll
  for (int o = 16; o; o >>= 1) s += __shfl_xor(s, o, 32);
  out[(size_t)row * 32 + lane] = v - m - __logf(s);
}

// ---------------------------------------------------------------------------
// One full GATConv layer (add_self_loops=True). out may alias x: x is only
// read by the leading GEMM, and all kernels are stream-ordered.
// ---------------------------------------------------------------------------
static void gat_layer(const float* x, const int* ei, int E, int N,
                      const float* W, const float* aS, const float* aD,
                      const float* bias, int Fin, int H, int C, bool relu,
                      float* hbuf, float* out, float* as_, float* ad_,
                      unsigned* mo, float* sb, float* alphab,
                      hipStream_t stream) {
  const int F = H * C;
  const int Hs = __builtin_ctz(H);
  const int Cs = __builtin_ctz(C);
  const int BLK = 256;
  {  // projection: hbuf[N,F] = x[N,Fin] @ W[Fin,F]
    if (F % 64 == 0) {
      int strips = (N >> 4) * (F / 64);
      wmma_gemm_f32<4><<<(strips * 32 + BLK - 1) / BLK, BLK, 0, stream>>>(
          x, W, hbuf, N, F, Fin);
    } else {  // F == 32
      int strips = (N >> 4) * (F / 32);
      wmma_gemm_f32<2><<<(strips * 32 + BLK - 1) / BLK, BLK, 0, stream>>>(
          x, W, hbuf, N, F, Fin);
    }
  }
  {  // attention dots
    int t = N * H;
    alpha_dots_k<<<(t + BLK - 1) / BLK, BLK, 0, stream>>>(hbuf, aS, aD, as_,
                                                          ad_, N, Hs, Cs);
  }
  {  // init max (ordered 0 == -inf), sum, and output accumulator
    size_t t = (size_t)N * H;
    fill_u32_k<<<(unsigned)((t + BLK - 1) / BLK), BLK, 0, stream>>>(mo, 0u, t);
    fill_u32_k<<<(unsigned)((t + BLK - 1) / BLK), BLK, 0, stream>>>(
        (unsigned*)sb, 0u, t);
    size_t to = (size_t)N * F;
    fill_u32_k<<<(unsigned)((to + BLK - 1) / BLK), BLK, 0, stream>>>(
        (unsigned*)out, 0u, to);
  }
  {  // softmax over dst neighborhoods
    size_t t = (size_t)(E + N) * H;
    unsigned g = (unsigned)((t + BLK - 1) / BLK);
    edge_max_k<<<g, BLK, 0, stream>>>(ei, E, N, Hs, as_, ad_, mo);
    edge_sum_k<<<g, BLK, 0, stream>>>(ei, E, N, Hs, as_, ad_, mo, sb);
    edge_alpha_k<<<g, BLK, 0, stream>>>(ei, E, N, Hs, as_, ad_, mo, sb,
                                        alphab);
  }
  {  // weighted aggregation
    size_t t = (size_t)(E + N) * F;
    edge_aggr_k<<<(unsigned)((t + BLK - 1) / BLK), BLK, 0, stream>>>(
        ei, E, N, Hs, Cs, alphab, hbuf, out);
  }
  {  // bias (+ReLU); heads=1/concat=False mean is identity
    size_t t = (size_t)N * F;
    bias_act_k<<<(unsigned)((t + BLK - 1) / BLK), BLK, 0, stream>>>(
        out, bias, F - 1, t, relu ? 1 : 0);
  }
}

extern "C" void kernel_launch(void* const* d_in, const int* in_sizes, int n_in,
                              void* d_out, int out_size, void* d_ws,
                              size_t ws_size, hipStream_t stream) {
  const float* dept_x   = (const float*)d_in[0];
  const int*   dept_ei  = (const int*)d_in[1];
  const float* emp_x    = (const float*)d_in[2];
  const int*   emp_ei   = (const int*)d_in[3];
  const int*   dept_idx = (const int*)d_in[4];
  const float* W_d1 = (const float*)d_in[5];
  const float* aS_d1 = (const float*)d_in[6];
  const float* aD_d1 = (const float*)d_in[7];
  const float* b_d1 = (const float*)d_in[8];
  const float* W_d2 = (const float*)d_in[9];
  const float* aS_d2 = (const float*)d_in[10];
  const float* aD_d2 = (const float*)d_in[11];
  const float* b_d2 = (const float*)d_in[12];
  const float* W_e1 = (const float*)d_in[13];
  const float* aS_e1 = (const float*)d_in[14];
  const float* aD_e1 = (const float*)d_in[15];
  const float* b_e1 = (const float*)d_in[16];
  const float* W_e2 = (const float*)d_in[17];
  const float* aS_e2 = (const float*)d_in[18];
  const float* aD_e2 = (const float*)d_in[19];
  const float* b_e2 = (const float*)d_in[20];

  const int IN = 128, DH = 64, EH = 64, OUT = 32, H = 4;
  const int N_D = in_sizes[0] / IN;
  const int E_D = in_sizes[1] / 2;
  const int N_E = in_sizes[2] / IN;
  const int E_E = in_sizes[3] / 2;

  // Bump allocator over d_ws (256B aligned slabs).
  char* wp = (char*)d_ws;
  auto alloc = [&](size_t bytes) -> void* {
    void* p = wp;
    wp += (bytes + 255) & ~(size_t)255;
    return p;
  };
  float* bufA = (float*)alloc((size_t)N_E * 4 * EH * sizeof(float));  // h
  float* bufB = (float*)alloc((size_t)N_E * 4 * EH * sizeof(float));  // x/out
  float* bufC = (float*)alloc((size_t)N_E * OUT * sizeof(float));     // logits
  float* dvec = (float*)alloc((size_t)N_D * DH * sizeof(float));      // dept emb
  float* as_ = (float*)alloc((size_t)N_E * H * sizeof(float));
  float* ad_ = (float*)alloc((size_t)N_E * H * sizeof(float));
  unsigned* mo = (unsigned*)alloc((size_t)N_E * H * sizeof(unsigned));
  float* sb = (float*)alloc((size_t)N_E * H * sizeof(float));
  float* alphab = (float*)alloc((size_t)(E_E + N_E) * H * sizeof(float));

  // ---- dept GAT 1: [N_D,128] -> relu -> [N_D,256] (4 heads, concat) ----
  gat_layer(dept_x, dept_ei, E_D, N_D, W_d1, aS_d1, aD_d1, b_d1, IN, H, DH,
            true, bufA, bufB, as_, ad_, mo, sb, alphab, stream);
  // ---- dept GAT 2: [N_D,256] -> [N_D,64] (1 head) ----
  gat_layer(bufB, dept_ei, E_D, N_D, W_d2, aS_d2, aD_d2, b_d2, H * DH, 1, DH,
            false, bufA, dvec, as_, ad_, mo, sb, alphab, stream);
  // ---- hierarchical fusion: e_in[N_E,192] = [emp_x | dvec[dept_idx]] ----
  {
    size_t t = (size_t)N_E * (IN + DH);
    concat_k<<<(unsigned)((t + 255) / 256), 256, 0, stream>>>(
        emp_x, dvec, dept_idx, bufB, N_E, IN, DH);
  }
  // ---- emp GAT 1: [N_E,192] -> relu -> [N_E,256] (4 heads, concat) ----
  gat_layer(bufB, emp_ei, E_E, N_E, W_e1, aS_e1, aD_e1, b_e1, IN + DH, H, EH,
            true, bufA, bufB, as_, ad_, mo, sb, alphab, stream);
  // ---- emp GAT 2: [N_E,256] -> [N_E,32] (1 head) ----
  gat_layer(bufB, emp_ei, E_E, N_E, W_e2, aS_e2, aD_e2, b_e2, H * EH, 1, OUT,
            false, bufA, bufC, as_, ad_, mo, sb, alphab, stream);
  // ---- log_softmax over 32 classes: one wave32 per row ----
  {
    size_t t = (size_t)N_E * 32;
    log_softmax32_k<<<(unsigned)((t + 255) / 256), 256, 0, stream>>>(
        bufC, (float*)d_out, N_E);
  }
}